// FeatureFuser_72533407695354
// MI455X (gfx1250) — compile-verified
//
#include <hip/hip_runtime.h>
#include <hip/hip_bf16.h>

// FeatureFuser: windowed overwrite of sampling_map by refined maps (last k
// wins), then sigmoid. Pure HBM-streaming kernel:
//   traffic floor = 64 MiB read + 64 MiB write -> ~5.8 us @ 23.3 TB/s.
// Round 2: 4 c-planes per thread (mask is c-invariant, amortize select math
// over 128 B of traffic) and sigmoid via gfx1250 v_tanh_f32 (3 VALU/elem).

typedef float __attribute__((ext_vector_type(4))) vf4;

#define GRIDSZ 4
#define WING   3
#define Bv     8
#define Kv     4
#define Cv     32
#define Hv     256
#define Wv     256
#define GH     (Hv / GRIDSZ)        // 64
#define GW     (Wv / GRIDSZ)        // 64
#define PLANE4 (Hv * Wv / 4)        // 16384 float4 per (b,c) plane
#define CSTEP  8                    // thread covers c, c+8, c+16, c+24

__device__ __forceinline__ float sigmoid_fast(float x) {
#if __has_builtin(__builtin_amdgcn_tanhf)
  // sigmoid(x) = 0.5*tanh(0.5*x) + 0.5  ->  v_mul, v_tanh_f32, v_fma (3 ops)
  return __builtin_fmaf(0.5f, __builtin_amdgcn_tanhf(0.5f * x), 0.5f);
#else
  // fallback: exp2 + rcp (4 ops)
  float e = __builtin_amdgcn_exp2f(-x * 1.4426950408889634f);
  return __builtin_amdgcn_rcpf(1.0f + e);
#endif
}

__device__ __forceinline__ vf4 sigmoid4(vf4 v) {
  vf4 r;
  r.x = sigmoid_fast(v.x);
  r.y = sigmoid_fast(v.y);
  r.z = sigmoid_fast(v.z);
  r.w = sigmoid_fast(v.w);
  return r;
}

__global__ __launch_bounds__(256) void FeatureFuser_72533407695354_kernel(
    const float* __restrict__ sampling,   // [B,C,H,W]
    const float* __restrict__ refined,    // [B,K,C,H,W]
    const int*   __restrict__ regions,    // [B,K,2] (y,x)
    float* __restrict__ out)              // [B,C,H,W]
{
  const unsigned blk = blockIdx.x;        // 0..4095
  const unsigned tid = threadIdx.x;       // 0..255

  // b, c0 derived from blockIdx only -> block-uniform -> scalar regs/s_load.
  const unsigned b  = blk >> 9;                            // 0..7
  const unsigned c0 = (blk >> 6) & 7u;                     // 0..7
  const unsigned h  = ((blk << 2) | (tid >> 6)) & 255u;    // 0..255
  const unsigned w4 = tid & 63u;                           // float4 column
  const int w  = (int)(w4 << 2);
  const int hi = (int)h;

  // 4 window descriptors for this b: block-uniform (single s_load_b256).
  const int* reg = regions + b * (Kv * 2);

  // Branchless "last k whose window contains (h,w)". Window x/y bounds are
  // multiples of 64, so the whole float4 shares one verdict.
  int sel = -1;
#pragma unroll
  for (int k = 0; k < Kv; ++k) {
    const int y0 = reg[2 * k] * GH;
    const int x0 = reg[2 * k + 1] * GW;
    const int y1 = min(y0 + WING * GH, Hv);
    const int x1 = min(x0 + WING * GW, Wv);
    const bool in = (hi >= y0) & (hi < y1) & (w >= x0) & (w < x1);
    sel = in ? k : sel;
  }

  const unsigned inner = ((h << 8) | (unsigned)w) >> 2;    // (h*W + w)/4

  // Exactly one source per output element.
  const vf4* src;
  unsigned off;
  if (sel < 0) {
    src = (const vf4*)sampling;
    off = (b * Cv + c0) * PLANE4 + inner;
  } else {
    src = (const vf4*)refined;
    off = ((b * Kv + (unsigned)sel) * Cv + c0) * PLANE4 + inner;
  }

  // Streamed-once data: non-temporal b128 (gfx1250 TH cache hints).
  const vf4 v0 = __builtin_nontemporal_load(src + off);
  const vf4 v1 = __builtin_nontemporal_load(src + off + 1u * CSTEP * PLANE4);
  const vf4 v2 = __builtin_nontemporal_load(src + off + 2u * CSTEP * PLANE4);
  const vf4 v3 = __builtin_nontemporal_load(src + off + 3u * CSTEP * PLANE4);

  const vf4 s0 = sigmoid4(v0);
  const vf4 s1 = sigmoid4(v1);
  const vf4 s2 = sigmoid4(v2);
  const vf4 s3 = sigmoid4(v3);

  vf4* o4 = (vf4*)out;
  const unsigned oo = (b * Cv + c0) * PLANE4 + inner;
  __builtin_nontemporal_store(s0, o4 + oo);
  __builtin_nontemporal_store(s1, o4 + oo + 1u * CSTEP * PLANE4);
  __builtin_nontemporal_store(s2, o4 + oo + 2u * CSTEP * PLANE4);
  __builtin_nontemporal_store(s3, o4 + oo + 3u * CSTEP * PLANE4);
}

extern "C" void kernel_launch(void* const* d_in, const int* in_sizes, int n_in,
                              void* d_out, int out_size, void* d_ws, size_t ws_size,
                              hipStream_t stream) {
  const float* sampling = (const float*)d_in[0];
  const float* refined  = (const float*)d_in[1];
  const int*   regions  = (const int*)d_in[2];
  float*       out      = (float*)d_out;

  // 8 b * 8 c-quarters * 256 h * 64 w4 = 1,048,576 threads = 4096 blocks x 256
  dim3 grid(4096), block(256);
  hipLaunchKernelGGL(FeatureFuser_72533407695354_kernel, grid, block, 0, stream,
                     sampling, refined, regions, out);
}